// LaplacianRegularizer_16295105921434
// MI455X (gfx1250) — compile-verified
//
#include <hip/hip_runtime.h>

// LaplacianRegularizer on MI455X (gfx1250).
// f: (4,12,1024,1024) fp32, 192MB. Memory-bound: HBM floor ~8.2us @ 23.3TB/s.
// Strategy: each 256-thread block streams a band of rows of one plane through
// LDS using async global->LDS b128 copies (ASYNCcnt pipeline, depth 2, 4 row
// buffers), computes the symmetry-reduced pair sums from LDS, then reduces
// per-wave with V_WMMA_F32_16X16X4_F32 (B = ones -> exact row-sum reduction),
// writes one partial per block, and a final kernel does a deterministic sum.

#define W 1024
#define H 1024
#define NPLANES 48          // 4 * 12
#define TPB 256             // 8 wave32 waves; 256*float4 = one full row
#define BANDS 16
#define BAND_ROWS 64        // ceil((H-1)/BANDS)
#define NBLOCKS (NPLANES * BANDS)
#define NBUF 4              // LDS row ring buffer (16KB)

typedef __attribute__((ext_vector_type(2))) float v2f;
typedef __attribute__((ext_vector_type(8))) float v8f;

// Exact pointee type the async-LDS builtin expects: int, vector_size(16).
typedef int v4i_vs __attribute__((vector_size(16)));
typedef __attribute__((address_space(1))) v4i_vs* as1_v4i_p;
typedef __attribute__((address_space(3))) v4i_vs* as3_v4i_p;

#if defined(__HIP_DEVICE_COMPILE__)
#if __has_builtin(__builtin_amdgcn_global_load_async_to_lds_b128)
#define HAVE_ASYNC_LDS 1
#endif
#if __has_builtin(__builtin_amdgcn_wmma_f32_16x16x4_f32)
#define HAVE_WMMA_F32X4 1
#endif
#if __has_builtin(__builtin_amdgcn_s_wait_asynccnt)
#define WAIT_ASYNC(n) __builtin_amdgcn_s_wait_asynccnt(n)
#else
#define WAIT_ASYNC(n) asm volatile("s_wait_asynccnt %0" ::"i"(n) : "memory")
#endif
#endif

__device__ __forceinline__ float sq(float x) { return x * x; }

// Stage 16 bytes (float4) of a row from global into LDS.
__device__ __forceinline__ void stage_row16(float* lds_dst, const float* gsrc) {
#if defined(HAVE_ASYNC_LDS)
  __builtin_amdgcn_global_load_async_to_lds_b128(
      (as1_v4i_p)gsrc, (as3_v4i_p)lds_dst, /*imm offset*/ 0, /*cpol*/ 0);
#else
  *(float4*)lds_dst = *(const float4*)gsrc;  // synchronous fallback
#endif
}

// Exact wave32 sum via one WMMA: D = A x Ones + 0. Every column of D equals
// the row-sums of A; lanes <16 hold rowsums of rows 0..7 in d[0..7], lanes
// >=16 hold rows 8..15. Requires EXEC all ones (all callers fully active).
__device__ __forceinline__ float wave_sum(float v) {
#if defined(HAVE_WMMA_F32X4)
  v2f a = {v, 0.0f};
  v2f b = {1.0f, 1.0f};
  v8f c = {0.f, 0.f, 0.f, 0.f, 0.f, 0.f, 0.f, 0.f};
  v8f d = __builtin_amdgcn_wmma_f32_16x16x4_f32(false, a, false, b,
                                                (short)0, c, false, false);
  float s = ((d[0] + d[1]) + (d[2] + d[3])) + ((d[4] + d[5]) + (d[6] + d[7]));
  s += __shfl_xor(s, 16, 32);  // rows 0..7 total + rows 8..15 total
  return s;
#else
  float s = v;
#pragma unroll
  for (int o = 16; o > 0; o >>= 1) s += __shfl_xor(s, o, 32);
  return s;
#endif
}

__global__ void __launch_bounds__(TPB)
lap_partial_kernel(const float* __restrict__ f, float* __restrict__ partial) {
  __shared__ float rows[NBUF][W];     // 16KB row ring buffer
  __shared__ float wsum[TPB / 32];

  const int t = threadIdx.x;
  const int col = t * 4;
  const int plane = blockIdx.x / BANDS;
  const int band = blockIdx.x % BANDS;
  const int i0 = band * BAND_ROWS;
  const int iend = (i0 + BAND_ROWS < H - 1) ? (i0 + BAND_ROWS) : (H - 1);
  const float* base = f + (size_t)plane * (size_t)(W * H);

  // Prime the pipeline: rows i0, i0+1, i0+2 (rows i0..iend are consumed).
  stage_row16(&rows[i0 & (NBUF - 1)][col], base + (size_t)i0 * W + col);
  stage_row16(&rows[(i0 + 1) & (NBUF - 1)][col], base + (size_t)(i0 + 1) * W + col);
  if (i0 + 2 <= iend)
    stage_row16(&rows[(i0 + 2) & (NBUF - 1)][col], base + (size_t)(i0 + 2) * W + col);

  const float wx = (t == 0) ? 2.0f : 1.0f;        // global col 0: vert weight 2
  const float ww = (t == TPB - 1) ? 2.0f : 1.0f;  // global col W-1: vert weight 2
  const bool hasR = (t < TPB - 1);
  const bool hasL = (t > 0);

  float acc = 0.0f;
  for (int i = i0; i < iend; ++i) {
#if defined(HAVE_ASYNC_LDS)
    // Rows <= i+1 must be complete; row i+2 (if issued) may stay in flight.
    if (i + 2 <= iend) { WAIT_ASYNC(1); } else { WAIT_ASYNC(0); }
#endif
    __syncthreads();  // all waves' row portions visible; prev compute retired
    if (i + 3 <= iend)  // overlap next-next row load with this compute
      stage_row16(&rows[(i + 3) & (NBUF - 1)][col], base + (size_t)(i + 3) * W + col);

    const float* cr = rows[i & (NBUF - 1)];
    const float* nr = rows[(i + 1) & (NBUF - 1)];
    float4 c = *(const float4*)(cr + col);
    float4 n = *(const float4*)(nr + col);
    float cRv = cr[(col + 4) & (W - 1)];
    float nRv = nr[(col + 4) & (W - 1)];
    float nLv = nr[(col - 1) & (W - 1)];

    // Horizontal pairs of row i (weight 2 on row 0).
    float hh = sq(c.x - c.y) + sq(c.y - c.z) + sq(c.z - c.w);
    if (hasR) hh += sq(c.w - cRv);
    // Vertical pairs (i,i+1); border columns get weight 2.
    float vv = wx * sq(c.x - n.x) + sq(c.y - n.y) + sq(c.z - n.z) + ww * sq(c.w - n.w);
    // Diagonal down-right.
    float dd = sq(c.x - n.y) + sq(c.y - n.z) + sq(c.z - n.w);
    if (hasR) dd += sq(c.w - nRv);
    // Anti-diagonal down-left.
    float aa = sq(c.y - n.x) + sq(c.z - n.y) + sq(c.w - n.z);
    if (hasL) aa += sq(c.x - nLv);

    float s = ((i == 0) ? 2.0f : 1.0f) * hh + vv + dd + aa;
    if (i + 1 == H - 1) {  // last row's horizontal pairs, weight 2
      float h2 = sq(n.x - n.y) + sq(n.y - n.z) + sq(n.z - n.w);
      if (hasR) h2 += sq(n.w - nRv);
      s += 2.0f * h2;
    }
    acc += s;
  }
  acc *= 2.0f;  // global pair-symmetry factor

  float s = wave_sum(acc);
  const int lane = t & 31, wave = t >> 5;
  if (lane == 0) wsum[wave] = s;
  __syncthreads();
  if (t == 0) {
    float bsum = 0.0f;
#pragma unroll
    for (int k = 0; k < TPB / 32; ++k) bsum += wsum[k];
    partial[blockIdx.x] = bsum;
  }
}

__global__ void __launch_bounds__(TPB)
lap_final_kernel(const float* __restrict__ partial, float* __restrict__ out) {
  __shared__ float wsum[TPB / 32];
  float v = 0.0f;
  for (int k = threadIdx.x; k < NBLOCKS; k += TPB) v += partial[k];
  float s = wave_sum(v);
  const int lane = threadIdx.x & 31, wave = threadIdx.x >> 5;
  if (lane == 0) wsum[wave] = s;
  __syncthreads();
  if (threadIdx.x == 0) {
    float bsum = 0.0f;
#pragma unroll
    for (int k = 0; k < TPB / 32; ++k) bsum += wsum[k];
    out[0] = bsum;  // deterministic fixed-order reduction, no atomics
  }
}

extern "C" void kernel_launch(void* const* d_in, const int* in_sizes, int n_in,
                              void* d_out, int out_size, void* d_ws, size_t ws_size,
                              hipStream_t stream) {
  const float* f = (const float*)d_in[0];
  float* partial = (float*)d_ws;  // NBLOCKS floats = 3KB scratch
  lap_partial_kernel<<<NBLOCKS, TPB, 0, stream>>>(f, partial);
  lap_final_kernel<<<1, TPB, 0, stream>>>(partial, (float*)d_out);
}